// VectorQuantizer_6889127543124
// MI455X (gfx1250) — compile-verified
//
#include <hip/hip_runtime.h>
#include <hip/hip_bf16.h>
#include <stdint.h>

// VQ-VAE vector quantizer for MI455X (gfx1250).
// z_e: (B=16, C=64, H=64, W=64) fp32; emb: (1024, 64) fp32.
// Outputs concatenated: z_q (4194304 f32) | loss (1 f32) | idx (65536 i32).
//
// Distance GEMM runs on V_WMMA_F32_16X16X4_F32 (exact fp32 matrix pipe):
// argmin over score = ||e||^2 - 2 z.e  (||z||^2 dropped: constant per row).
// Codebook chunks are double-buffered into LDS via GLOBAL_LOAD_ASYNC_TO_LDS_B128
// (ASYNCcnt-tracked, no VGPR round trip) so staging overlaps the WMMA sweep.

typedef float v2f __attribute__((ext_vector_type(2)));
typedef float v8f __attribute__((ext_vector_type(8)));

#define N_EMBED   1024
#define EMBED_DIM 64
#define HW        4096          // 64*64
#define N_ROWS    65536         // 16*64*64
#define CHUNK     128           // codebook rows staged in LDS per step
#define N_CHUNKS  (N_EMBED / CHUNK)            // 8
#define LDS_STRIDE 68           // 64 + 4 pad floats; 272 B row = 16B aligned,
                                // lanes hit distinct banks on B-fragment reads
#define WAVES_PER_BLOCK 8
#define BLOCK_THREADS 256
#define ROWS_PER_BLOCK (WAVES_PER_BLOCK * 16)
#define NUM_BLOCKS (N_ROWS / ROWS_PER_BLOCK)   // 512

// Issue one chunk (128 codebook rows, 32 KB) of async global->LDS traffic.
// GVS addressing: saddr = emb base (SGPR pair), vaddr = 32-bit byte offset,
// vdst = wave-relative LDS byte offset. 8 x B128 per thread.
__device__ __forceinline__ void stage_chunk_async(const float* __restrict__ emb,
                                                  int chunkRow, float* buf, int tid)
{
    const unsigned lds_base = (unsigned)(uintptr_t)buf;       // low 32b = LDS offset
    const uint64_t gbase    = (uint64_t)(uintptr_t)emb;
#pragma unroll
    for (int it = 0; it < (CHUNK * EMBED_DIM / 4) / BLOCK_THREADS; ++it) { // 8
        const int i  = tid + it * BLOCK_THREADS;
        const int rr = i >> 4;                 // 16 float4 per codebook row
        const int cc = (i & 15) << 2;
        const unsigned lds_off = lds_base + (unsigned)(rr * LDS_STRIDE + cc) * 4u;
        const unsigned gl_off  = (unsigned)(((chunkRow + rr) * EMBED_DIM + cc) * 4);
        asm volatile("global_load_async_to_lds_b128 %0, %1, %2"
                     :: "v"(lds_off), "v"(gl_off), "s"(gbase) : "memory");
    }
}

__device__ __forceinline__ void wait_async0()
{
    asm volatile("s_wait_asynccnt 0x0" ::: "memory");
}

__global__ __launch_bounds__(BLOCK_THREADS)
void vq_main_kernel(const float* __restrict__ ze,
                    const float* __restrict__ emb,
                    float* __restrict__ out_zq,
                    int*   __restrict__ out_idx,
                    float* __restrict__ ws_partials)
{
    __shared__ __align__(16) float lds_tile[2][CHUNK * LDS_STRIDE]; // ~70 KB total
    __shared__ float lds_enorm[2][CHUNK];
    __shared__ float lds_wsum[WAVES_PER_BLOCK];

    const int tid  = threadIdx.x;
    const int wave = tid >> 5;
    const int lane = tid & 31;
    const int l16  = lane & 15;
    const int half = lane >> 4;          // 0: M rows 0-7 / K 0-1; 1: M rows 8-15 / K 2-3
    const int koff = 2 * half;

    // Kick chunk 0 into buffer 0 before anything else; A-fragment loads overlap it.
    stage_chunk_async(emb, 0, &lds_tile[0][0], tid);

    // 16 consecutive z-rows per wave; never crosses a batch image (128 | 4096).
    const int rowBase = blockIdx.x * ROWS_PER_BLOCK + wave * 16;
    const int b       = rowBase >> 12;       // rowBase / 4096
    const int hwBase  = rowBase & 4095;
    const float* zb   = ze + (size_t)b * EMBED_DIM * HW;

    // ---- A fragments: z tile 16x64 (fp32), held for the whole sweep.
    // ISA layout (32-bit A 16x4): lane = M (mod 16); VGPR j holds K = j + 2*(lane/16).
    v2f za[16];
#pragma unroll
    for (int s = 0; s < 16; ++s) {
        const int k = 4 * s + koff;
        za[s].x = zb[(size_t)(k    ) * HW + hwBase + l16];
        za[s].y = zb[(size_t)(k + 1) * HW + hwBase + l16];
    }

    float best[8];
    int   bestIdx[8];
#pragma unroll
    for (int r = 0; r < 8; ++r) { best[r] = 3.4e38f; bestIdx[r] = 0; }

    // Hand over buffer 0: my async loads done -> barrier -> ||e||^2 -> barrier.
    wait_async0();
    __syncthreads();
    if (tid < CHUNK) {
        const float* rowp = &lds_tile[0][tid * LDS_STRIDE];
        float s = 0.f;
#pragma unroll
        for (int c2 = 0; c2 < EMBED_DIM; ++c2) s += rowp[c2] * rowp[c2];
        lds_enorm[0][tid] = s;
    }
    __syncthreads();

    int p = 0;
    for (int c = 0; c < N_CHUNKS; ++c) {
        // Prefetch next chunk into the spare buffer while we compute on this one.
        if (c + 1 < N_CHUNKS)
            stage_chunk_async(emb, (c + 1) * CHUNK, &lds_tile[p ^ 1][0], tid);

        const float* tile = &lds_tile[p][0];
        const float* enp  = &lds_enorm[p][0];
        const int chunkBase = c * CHUNK;

#pragma unroll
        for (int g = 0; g < CHUNK / 16; ++g) {
            v8f acc = {};
            // B fragment (4x16 = emb^T): lane = N (mod 16); VGPR j holds K = j + 2*(lane/16).
            const float* bp = &tile[(g * 16 + l16) * LDS_STRIDE + koff];
#pragma unroll
            for (int s = 0; s < 16; ++s) {     // K = 64 in 16 steps of 4
                v2f bf;
                bf.x = bp[4 * s];
                bf.y = bp[4 * s + 1];
                acc = __builtin_amdgcn_wmma_f32_16x16x4_f32(
                    false, za[s], false, bf, (short)0, acc, false, false);
            }
            // D layout: VGPR r holds D[r + 8*half][l16]. Running argmin per lane.
            const int   col = chunkBase + g * 16 + l16;
            const float en  = enp[g * 16 + l16];
#pragma unroll
            for (int r = 0; r < 8; ++r) {
                const float d = en - 2.0f * acc[r];
                if (d < best[r] || (d == best[r] && col < bestIdx[r])) {
                    best[r] = d; bestIdx[r] = col;
                }
            }
        }

        if (c + 1 < N_CHUNKS) {
            wait_async0();       // my prefetch landed
            __syncthreads();     // everyone's landed; everyone done reading tile[p]
            if (tid < CHUNK) {   // ||e||^2 for the incoming buffer
                const float* rowp = &lds_tile[p ^ 1][tid * LDS_STRIDE];
                float s = 0.f;
#pragma unroll
                for (int c2 = 0; c2 < EMBED_DIM; ++c2) s += rowp[c2] * rowp[c2];
                lds_enorm[p ^ 1][tid] = s;
            }
            __syncthreads();
        }
        p ^= 1;
    }

    // ---- Butterfly reduce over the 16 lanes that share each row (tie: lowest idx,
    // matching jnp.argmin "first minimum" semantics).
#pragma unroll
    for (int r = 0; r < 8; ++r) {
#pragma unroll
        for (int m = 8; m >= 1; m >>= 1) {
            const float ov = __shfl_xor(best[r], m, 16);
            const int   oi = __shfl_xor(bestIdx[r], m, 16);
            if (ov < best[r] || (ov == best[r] && oi < bestIdx[r])) {
                best[r] = ov; bestIdx[r] = oi;
            }
        }
    }
    if (l16 == 0) {        // lane 0 -> rows rowBase+0..7, lane 16 -> rows rowBase+8..15
#pragma unroll
        for (int r = 0; r < 8; ++r)
            out_idx[rowBase + half * 8 + r] = bestIdx[r];
    }

    // ---- Gather z_q = emb[idx], write in (B,C,H,W) layout, accumulate loss.
    float lsum = 0.f;
#pragma unroll
    for (int r2 = 0; r2 < 16; ++r2) {
        const int src = (r2 < 8) ? 0 : 16;
        const int bi  = __shfl(bestIdx[r2 & 7], src, 32);
        const int hw  = hwBase + r2;
        const float* er = emb    + (size_t)bi * EMBED_DIM;
        float*       ob = out_zq + (size_t)b * EMBED_DIM * HW + hw;
        const float* zr = zb + hw;
#pragma unroll
        for (int cc = 0; cc < 2; ++cc) {
            const int c = lane + cc * 32;        // 32 lanes x 2 cover C=64
            const float q = er[c];               // coalesced
            const float z = zr[(size_t)c * HW];  // strided; sequential over r2 per lane
            ob[(size_t)c * HW] = q;
            const float df = q - z;
            lsum += df * df;
        }
    }

    // Deterministic block reduction (fixed order), per-block partial to ws.
#pragma unroll
    for (int m = 16; m >= 1; m >>= 1) lsum += __shfl_xor(lsum, m, 32);
    if (lane == 0) lds_wsum[wave] = lsum;
    __syncthreads();
    if (tid == 0) {
        float s = 0.f;
        for (int wv = 0; wv < WAVES_PER_BLOCK; ++wv) s += lds_wsum[wv];
        ws_partials[blockIdx.x] = s;
    }
}

__global__ void vq_finalize_kernel(const float* __restrict__ ws_partials,
                                   float* __restrict__ out_loss)
{
    if (threadIdx.x == 0 && blockIdx.x == 0) {
        float s = 0.f;
        for (int i = 0; i < NUM_BLOCKS; ++i) s += ws_partials[i];  // fixed order
        // codebook_loss + beta*commit_loss; both equal mean((z_q - z_e)^2) forward.
        *out_loss = (1.0f + 0.25f) * s / (float)(N_ROWS * EMBED_DIM);
    }
}

extern "C" void kernel_launch(void* const* d_in, const int* in_sizes, int n_in,
                              void* d_out, int out_size, void* d_ws, size_t ws_size,
                              hipStream_t stream) {
    const float* ze  = (const float*)d_in[0];   // (16, 64, 64, 64) fp32
    const float* emb = (const float*)d_in[1];   // (1024, 64) fp32

    float* out      = (float*)d_out;
    float* out_zq   = out;                       // 4194304 floats (B,C,H,W)
    float* out_loss = out + (size_t)N_ROWS * EMBED_DIM;             // scalar
    int*   out_idx  = (int*)(out + (size_t)N_ROWS * EMBED_DIM + 1); // 65536 i32

    float* partials = (float*)d_ws;              // NUM_BLOCKS floats

    vq_main_kernel<<<dim3(NUM_BLOCKS), dim3(BLOCK_THREADS), 0, stream>>>(
        ze, emb, out_zq, out_idx, partials);
    vq_finalize_kernel<<<dim3(1), dim3(32), 0, stream>>>(partials, out_loss);
}